// FrequencyAdaptiveNorm_14353780704082
// MI455X (gfx1250) — compile-verified
//
#include <hip/hip_runtime.h>
#include <math.h>

// Problem constants (match reference setup_inputs)
#define B_      32
#define L_      4096
#define F_      128
#define TILE_L  128
#define HALO_L  10               // w=20 needs p-10 at tile start
#define HALO_R  9                // w=20 needs p+9 at tile end
#define ROWS    (TILE_L + HALO_L + HALO_R)   // 147 rows
#define NSEG    (L_ / TILE_L)                // 32 segments
#define EPS_    1e-5f

typedef float v2f __attribute__((ext_vector_type(2)));
typedef float v8f __attribute__((ext_vector_type(8)));

__device__ __forceinline__ float san(float v) {
    // nan_to_num(nan=0, posinf=0, neginf=0)
    return __builtin_isfinite(v) ? v : 0.0f;
}

// CDNA5 async DMA: copies 16B/lane from the global address saddr plus goff to the LDS offset in loff, tracked by ASYNCcnt.
__device__ __forceinline__ void async_ld_b128(unsigned loff, unsigned goff,
                                              const float* saddr) {
    asm volatile("global_load_async_to_lds_b128 %0, %1, %2"
                 :
                 : "v"(loff), "v"(goff), "s"(saddr)
                 : "memory");
}

__global__ __launch_bounds__(F_)
void FrequencyAdaptiveNorm_kernel(const float* __restrict__ x,
                                  const float* __restrict__ w,
                                  float* __restrict__ out)
{
    __shared__ float tile[ROWS * F_];   // 147*128*4 = 75264 B
    __shared__ float wsum[4][2][32];    // [wave][sum|sumsq][feature-in-wave], 1 KB

    const int seg = blockIdx.x;         // L segment
    const int b   = blockIdx.y;         // batch
    const int p0  = seg * TILE_L;
    const int tid = (int)threadIdx.x;   // 0..127, tid == feature index
    const int wv  = tid >> 5;           // wave id 0..3 (wave32)
    const int ln  = tid & 31;           // lane id

    // Interior segments: all halo rows in [0,L) AND all p valid for all windows.
    const bool interior = (seg != 0) && (seg != NSEG - 1);

    // ---------------- stage x[b, p0-10 .. p0+TILE_L+8, :] into LDS -------------
    // One wave per row per iteration: 32 lanes x b128 (16B) = 512B = one full row.
    {
        const unsigned lbase = (unsigned)(ln * 16);                  // bytes in row
        const unsigned gseg0 = (unsigned)(((b * L_ + (p0 - HALO_L)) * F_) * 4);
        if (interior) {
            for (int r = wv; r < ROWS; r += 4) {
                async_ld_b128((unsigned)(r * F_ * 4) + lbase,
                              gseg0 + (unsigned)(r * F_ * 4) + lbase, x);
            }
        } else {
            for (int r = wv; r < ROWS; r += 4) {
                const int grow = p0 - HALO_L + r;
                const unsigned loff = (unsigned)(r * F_ * 4) + lbase;
                if (grow >= 0 && grow < L_) {
                    async_ld_b128(loff,
                                  (unsigned)(((b * L_ + grow) * F_) * 4) + lbase, x);
                } else {
                    float4 z = make_float4(0.f, 0.f, 0.f, 0.f);
                    *reinterpret_cast<float4*>(&tile[r * F_ + ln * 4]) = z;
                }
            }
        }
    }
    asm volatile("s_wait_asynccnt 0x0" ::: "memory");  // my DMA writes landed
    __syncthreads();                                   // everyone's writes visible

    // ---------------- softmax over the 3 window weights ------------------------
    const float w0 = w[0], w1 = w[1], w2 = w[2];
    const float m  = fmaxf(w0, fmaxf(w1, w2));
    const float e0 = __expf(w0 - m), e1 = __expf(w1 - m), e2 = __expf(w2 - m);
    const float inv = 1.0f / (e0 + e1 + e2);
    const float ws5 = e0 * inv, ws10 = e1 * inv, ws20 = e2 * inv;

    // ---------------- w=20 init sums via WMMA row-reductions -------------------
    // With B = all-ones (layout-invariant: every slot is 1.0), v_wmma_f32_16x16x4_f32
    // yields D[m][n] = sum_k A[m][k] + C[m][n]  == fp32 row sums of A.
    // A layout (documented): lane l holds (M = l&15, K = (l>>4)*2 + {0,1}) in v0/v1.
    // 20 rows (p0-10 .. p0+9) = 5 chained K-chunks of 4. Two 16-feature groups/wave.
    {
        const int   mrow   = ln & 15;              // M row handled by this lane
        const int   cb     = (ln >> 4) * 2;        // this lane's K slots in a chunk
        const int   colLo  = (wv << 5) + mrow;     // feature column, low group
        const int   colHi  = colLo + 16;           // feature column, high group
        const v2f   bone   = {1.0f, 1.0f};
        v8f dSlo = {0,0,0,0,0,0,0,0}, dQlo = {0,0,0,0,0,0,0,0};
        v8f dShi = {0,0,0,0,0,0,0,0}, dQhi = {0,0,0,0,0,0,0,0};
#pragma unroll
        for (int i = 0; i < 5; ++i) {
            const int r0 = 4 * i + cb;             // local tile rows r0, r0+1
            const float xl0 = san(tile[(r0    ) * F_ + colLo]);
            const float xl1 = san(tile[(r0 + 1) * F_ + colLo]);
            const float xh0 = san(tile[(r0    ) * F_ + colHi]);
            const float xh1 = san(tile[(r0 + 1) * F_ + colHi]);
            v2f a;  a.x = xl0;        a.y = xl1;
            v2f aq; aq.x = xl0 * xl0; aq.y = xl1 * xl1;
            dSlo = __builtin_amdgcn_wmma_f32_16x16x4_f32(false, a,  false, bone,
                                                         (short)0, dSlo, false, false);
            dQlo = __builtin_amdgcn_wmma_f32_16x16x4_f32(false, aq, false, bone,
                                                         (short)0, dQlo, false, false);
            a.x = xh0;        a.y = xh1;
            aq.x = xh0 * xh0; aq.y = xh1 * xh1;
            dShi = __builtin_amdgcn_wmma_f32_16x16x4_f32(false, a,  false, bone,
                                                         (short)0, dShi, false, false);
            dQhi = __builtin_amdgcn_wmma_f32_16x16x4_f32(false, aq, false, bone,
                                                         (short)0, dQhi, false, false);
        }
        // D layout (documented): VGPR r, lanes 0-15 -> (M=r, N=lane);
        // lanes 16-31 -> (M=r+8). Row sums are replicated over N, so lanes
        // N==0 (lane 0) and N==0-of-upper-half (lane 16) write them back.
        if (ln == 0 || ln == 16) {
            const int mb = (ln == 0) ? 0 : 8;
#pragma unroll
            for (int r = 0; r < 8; ++r) {
                wsum[wv][0][mb + r]      = dSlo[r];
                wsum[wv][1][mb + r]      = dQlo[r];
                wsum[wv][0][16 + mb + r] = dShi[r];
                wsum[wv][1][16 + mb + r] = dQhi[r];
            }
        }
    }
    __syncthreads();

    // ---------------- streaming window sums along L ----------------------------
    const int f = tid;
    const float* tcol = &tile[f];       // per-thread column base
    float s20 = wsum[wv][0][ln];        // WMMA-computed init (p = p0)
    float q20 = wsum[wv][1][ln];
    float s5 = 0.f, q5 = 0.f, s10 = 0.f, q10 = 0.f;
#pragma unroll
    for (int k = -5; k <= 4; ++k) {     // scalar init for the small windows
        const float v = san(tcol[(HALO_L + k) * F_]);
        s10 += v; q10 += v * v;
        if (k >= -2 && k <= 2) { s5 += v; q5 += v * v; }
    }

    float* op = out + ((size_t)(b * L_ + p0)) * F_ + f;

    if (interior) {
        // Hot path: all windows valid at every p -> no per-step masks.
        for (int j = 0; j < TILE_L; ++j) {
            const int base = (j + HALO_L) * F_;
            if (j > 0) {
                const float a5  = san(tcol[base + 2  * F_]);
                const float d5  = san(tcol[base - 3  * F_]);
                const float a10 = san(tcol[base + 4  * F_]);
                const float d10 = san(tcol[base - 6  * F_]);
                const float a20 = san(tcol[base + 9  * F_]);
                const float d20 = san(tcol[base - 11 * F_]);
                s5  += a5  - d5;   q5  += a5  * a5  - d5  * d5;
                s10 += a10 - d10;  q10 += a10 * a10 - d10 * d10;
                s20 += a20 - d20;  q20 += a20 * a20 - d20 * d20;
            }
            const float xc = san(tcol[base]);

            const float mu5  = s5  * 0.2f;
            const float mu10 = s10 * 0.1f;
            const float mu20 = s20 * 0.05f;
            const float v5   = fmaxf(q5  * 0.2f  - mu5  * mu5,  0.0f);
            const float v10  = fmaxf(q10 * 0.1f  - mu10 * mu10, 0.0f);
            const float v20  = fmaxf(q20 * 0.05f - mu20 * mu20, 0.0f);
            float acc = ws5  * (xc - mu5)  * rsqrtf(v5  + EPS_)
                      + ws10 * (xc - mu10) * rsqrtf(v10 + EPS_)
                      + ws20 * (xc - mu20) * rsqrtf(v20 + EPS_);
            op[(size_t)j * F_] = acc;                 // coalesced 512B row / step
        }
    } else {
        // Edge segments: per-p validity masks (zero-padded output regions).
        for (int j = 0; j < TILE_L; ++j) {
            const int p    = p0 + j;
            const int base = (j + HALO_L) * F_;
            if (j > 0) {
                const float a5  = san(tcol[base + 2  * F_]);
                const float d5  = san(tcol[base - 3  * F_]);
                const float a10 = san(tcol[base + 4  * F_]);
                const float d10 = san(tcol[base - 6  * F_]);
                const float a20 = san(tcol[base + 9  * F_]);
                const float d20 = san(tcol[base - 11 * F_]);
                s5  += a5  - d5;   q5  += a5  * a5  - d5  * d5;
                s10 += a10 - d10;  q10 += a10 * a10 - d10 * d10;
                s20 += a20 - d20;  q20 += a20 * a20 - d20 * d20;
            }
            const float xc = san(tcol[base]);

            float acc = 0.0f;
            if (p >= 2 && p <= L_ - 5 + 2) {          // w=5 valid: [2, 4093]
                const float mu  = s5 * 0.2f;
                const float var = fmaxf(q5 * 0.2f - mu * mu, 0.0f);
                acc += ws5 * (xc - mu) * rsqrtf(var + EPS_);
            }
            if (p >= 5 && p <= L_ - 10 + 5) {         // w=10 valid: [5, 4091]
                const float mu  = s10 * 0.1f;
                const float var = fmaxf(q10 * 0.1f - mu * mu, 0.0f);
                acc += ws10 * (xc - mu) * rsqrtf(var + EPS_);
            }
            if (p >= 10 && p <= L_ - 20 + 10) {       // w=20 valid: [10, 4086]
                const float mu  = s20 * 0.05f;
                const float var = fmaxf(q20 * 0.05f - mu * mu, 0.0f);
                acc += ws20 * (xc - mu) * rsqrtf(var + EPS_);
            }
            op[(size_t)j * F_] = acc;
        }
    }
}

extern "C" void kernel_launch(void* const* d_in, const int* in_sizes, int n_in,
                              void* d_out, int out_size, void* d_ws, size_t ws_size,
                              hipStream_t stream) {
    (void)in_sizes; (void)n_in; (void)d_ws; (void)ws_size; (void)out_size;
    const float* x  = (const float*)d_in[0];   // (32, 4096, 128) fp32
    const float* wt = (const float*)d_in[1];   // (3,) fp32
    float* out      = (float*)d_out;           // (32, 4096, 128) fp32

    dim3 grid(NSEG, B_);                       // 32 segments x 32 batches
    dim3 block(F_);                            // 128 threads = 4 wave32
    FrequencyAdaptiveNorm_kernel<<<grid, block, 0, stream>>>(x, wt, out);
}